// ScoreNetworkGNN_67095979098913
// MI455X (gfx1250) — compile-verified
//
#include <hip/hip_runtime.h>
#include <hip/hip_bf16.h>

// Score-network GNN for MI455X (gfx1250), fp32 WMMA (16x16x4) throughout.
// Grid: one workgroup per batch (B=8), 256 threads = 8 wave32 waves.
// All intermediates resident in LDS (4 x 128x132 fp32 buffers ~ 270KB < 320KB/WGP).

typedef __attribute__((ext_vector_type(2))) float v2f;
typedef __attribute__((ext_vector_type(8))) float v8f;

#define NN   128   // nodes
#define HH   128   // hidden
#define DD   64    // node feature dim
#define LDST 132   // padded LDS row stride (bank-conflict-free strided reads)
#define EPSL 1e-5f

struct LayerP {
  const float *wmsg, *bmsg, *wupd, *bupd, *wedg, *bedg;
  const float *lnng, *lnnb, *lneg, *lneb;
};
struct NetP {
  const float *x, *a, *t;
  const float *w_t1, *b_t1, *w_t2, *b_t2;
  const float *w_np, *b_np, *w_ap, *b_ap;
  const float *w_sx1, *b_sx1, *w_sx2, *b_sx2;
  const float *w_sa1, *b_sa1, *w_sa2, *b_sa2;
  LayerP lyr[4];
  float *out_x;  // (B,N,D)
  float *u;      // workspace (B,N,H) = h_node @ W_sa1
};

__device__ __forceinline__ float silu_f(float x) {
  return x / (1.0f + __expf(-x));
}

// Accumulate C += A(row-block) * W over K, A has leading dim lda, W leading dim ldw.
// row/col/half2 are precomputed per-lane indices following the CDNA5 f32 WMMA layout:
//   A frag: lanes0-15 -> K = k0+{0,1}; lanes16-31 -> K = k0+{2,3}  (half2 = 2*(lane>>4))
//   B frag: VGPR0 -> rows k0 / k0+2, VGPR1 -> rows k0+1 / k0+3
__device__ __forceinline__ v8f wmma_span(const float* __restrict__ As, int lda, int row,
                                         const float* __restrict__ W, int ldw, int col,
                                         int K, int half2, v8f acc) {
#pragma unroll 4
  for (int k0 = 0; k0 < K; k0 += 4) {
    int ka = k0 + half2;
    v2f av, bv;
    av.x = As[row * lda + ka];
    av.y = As[row * lda + ka + 1];
    bv.x = W[ka * ldw + col];
    bv.y = W[(ka + 1) * ldw + col];
    acc = __builtin_amdgcn_wmma_f32_16x16x4_f32(false, av, false, bv,
                                                (short)0, acc, false, false);
  }
  return acc;
}

// Store 16x16 C tile: VGPR r holds rows (mt*16 + r) [lanes 0-15] / (mt*16 + r + 8) [lanes 16-31]
__device__ __forceinline__ void store_tile(float* dst, int ldd, int mt, int col,
                                           int half, v8f acc) {
  int rbase = mt * 16 + half * 8;
#pragma unroll
  for (int r = 0; r < 8; ++r) dst[(rbase + r) * ldd + col] = acc[r];
}

__global__ __launch_bounds__(256) void gnn_main(NetP p) {
  __shared__ float hn[NN * LDST];   // h_node
  __shared__ float he[NN * LDST];   // h_edge
  __shared__ float s1[NN * LDST];   // scratch (m / h_new / silu buf)
  __shared__ float s2[NN * LDST];   // scratch (x staging / msg)
  __shared__ float temb[HH];
  __shared__ float invden[NN];
  __shared__ float vcol[HH];        // per-column vector (te3 / edge mean term)
  __shared__ float meanh[HH];

  const int b    = blockIdx.x;
  const int tid  = threadIdx.x;
  const int wv   = tid >> 5;
  const int lane = tid & 31;
  const int half = lane >> 4;
  const int half2 = half * 2;
  const int l16  = lane & 15;
  const int mt   = wv;              // wave owns tile-row mt (M=128 -> 8 tile rows)
  const int arow = mt * 16 + l16;   // A-fragment row for this lane

  const float* aG = p.a + b * NN * NN;

  // ---- Stage 0: time-MLP hidden, stage x into LDS, row inv-denominators ----
  if (tid < HH) {
    float tv = p.t[b];
    float u0 = tv * p.w_t1[tid] + p.b_t1[tid];
    meanh[tid] = silu_f(u0);        // temp: silu(time1(t))
  }
  for (int idx = tid; idx < NN * DD; idx += 256) {
    s2[(idx / DD) * LDST + (idx % DD)] = p.x[b * NN * DD + idx];
  }
  if (tid < NN) {
    float s = 0.f;
    for (int j = 0; j < NN; ++j) s += fabsf(aG[tid * NN + j]);
    invden[tid] = 1.f / fmaxf(s, 1.f);
  }
  __syncthreads();
  if (tid < HH) {
    float acc = p.b_t2[tid];
    for (int k = 0; k < HH; ++k) acc += meanh[k] * p.w_t2[k * HH + tid];
    temb[tid] = acc;
  }
  __syncthreads();

  // ---- Stage 1: h_node = x @ W_np + b_np  (M=128,K=64,N=128), A in s2 ----
  for (int nt = 0; nt < 8; ++nt) {
    int col = nt * 16 + l16;
    v8f acc; float bb = p.b_np[col];
#pragma unroll
    for (int r = 0; r < 8; ++r) acc[r] = bb;
    acc = wmma_span(s2, LDST, arow, p.w_np, HH, col, DD, half2, acc);
    store_tile(hn, LDST, mt, col, half, acc);
  }
  // ---- Stage 2: h_edge = a @ W_ap + b_ap  (A from global) ----
  for (int nt = 0; nt < 8; ++nt) {
    int col = nt * 16 + l16;
    v8f acc; float bb = p.b_ap[col];
#pragma unroll
    for (int r = 0; r < 8; ++r) acc[r] = bb;
    acc = wmma_span(aG, NN, arow, p.w_ap, HH, col, NN, half2, acc);
    store_tile(he, LDST, mt, col, half, acc);
  }
  __syncthreads();

  // ---- Stage 3: message-passing layers ----
  for (int l = 0; l < 4; ++l) {
    const LayerP& L = p.lyr[l];

    // te3[h] = sum_k temb[k] * Wupd[256+k][h]   (time slice of node_update)
    if (tid < HH) {
      float acc = 0.f;
      for (int k = 0; k < HH; ++k) acc += temb[k] * L.wupd[(256 + k) * HH + tid];
      vcol[tid] = acc;
    }
    // m = hn @ Wmsg + bmsg -> s1
    for (int nt = 0; nt < 8; ++nt) {
      int col = nt * 16 + l16;
      v8f acc; float bb = L.bmsg[col];
#pragma unroll
      for (int r = 0; r < 8; ++r) acc[r] = bb;
      acc = wmma_span(hn, LDST, arow, L.wmsg, HH, col, HH, half2, acc);
      store_tile(s1, LDST, mt, col, half, acc);
    }
    __syncthreads();

    // msg = a_norm @ m -> s2   (A from global scaled by invden[row], B = s1 in LDS)
    {
      float iv = invden[arow];
      for (int nt = 0; nt < 8; ++nt) {
        int col = nt * 16 + l16;
        v8f acc;
#pragma unroll
        for (int r = 0; r < 8; ++r) acc[r] = 0.f;
#pragma unroll 4
        for (int k0 = 0; k0 < NN; k0 += 4) {
          int ka = k0 + half2;
          v2f av, bv;
          av.x = aG[arow * NN + ka] * iv;
          av.y = aG[arow * NN + ka + 1] * iv;
          bv.x = s1[ka * LDST + col];
          bv.y = s1[(ka + 1) * LDST + col];
          acc = __builtin_amdgcn_wmma_f32_16x16x4_f32(false, av, false, bv,
                                                      (short)0, acc, false, false);
        }
        store_tile(s2, LDST, mt, col, half, acc);
      }
    }
    __syncthreads();

    // h_node_new = hn@Wu1 + msg@Wu2 + te3 + bupd -> s1
    for (int nt = 0; nt < 8; ++nt) {
      int col = nt * 16 + l16;
      v8f acc; float bb = L.bupd[col] + vcol[col];
#pragma unroll
      for (int r = 0; r < 8; ++r) acc[r] = bb;
      acc = wmma_span(hn, LDST, arow, L.wupd, HH, col, HH, half2, acc);
      acc = wmma_span(s2, LDST, arow, L.wupd + 128 * HH, HH, col, HH, half2, acc);
      store_tile(s1, LDST, mt, col, half, acc);
    }
    __syncthreads();

    // node residual + LayerNorm -> hn
    if (tid < NN) {
      float mu = 0.f;
      for (int h = 0; h < HH; ++h) mu += hn[tid * LDST + h] + s1[tid * LDST + h];
      mu *= (1.f / HH);
      float var = 0.f;
      for (int h = 0; h < HH; ++h) {
        float v = hn[tid * LDST + h] + s1[tid * LDST + h] - mu;
        var += v * v;
      }
      var *= (1.f / HH);
      float rs = rsqrtf(var + EPSL);
      for (int h = 0; h < HH; ++h) {
        float v = hn[tid * LDST + h] + s1[tid * LDST + h] - mu;
        hn[tid * LDST + h] = v * rs * L.lnng[h] + L.lnnb[h];
      }
    }
    __syncthreads();

    // meanh[h] = mean_i hn[i][h]
    if (tid < HH) {
      float s = 0.f;
      for (int i = 0; i < NN; ++i) s += hn[i * LDST + tid];
      meanh[tid] = s * (1.f / NN);
    }
    __syncthreads();

    // edge mean term: vcol[h] = bedg[h] + sum_k meanh[k] * We2[k][h]
    if (tid < HH) {
      float acc = L.bedg[tid];
      for (int k = 0; k < HH; ++k) acc += meanh[k] * L.wedg[(128 + k) * HH + tid];
      vcol[tid] = acc;
    }
    __syncthreads();

    // h_edge_new = hn@We1 + he@We3 + vcol -> s1   (mean_j of the pair-MLP, collapsed)
    for (int nt = 0; nt < 8; ++nt) {
      int col = nt * 16 + l16;
      v8f acc; float bb = vcol[col];
#pragma unroll
      for (int r = 0; r < 8; ++r) acc[r] = bb;
      acc = wmma_span(hn, LDST, arow, L.wedg, HH, col, HH, half2, acc);
      acc = wmma_span(he, LDST, arow, L.wedg + 256 * HH, HH, col, HH, half2, acc);
      store_tile(s1, LDST, mt, col, half, acc);
    }
    __syncthreads();

    // edge residual + LayerNorm -> he
    if (tid < NN) {
      float mu = 0.f;
      for (int h = 0; h < HH; ++h) mu += he[tid * LDST + h] + s1[tid * LDST + h];
      mu *= (1.f / HH);
      float var = 0.f;
      for (int h = 0; h < HH; ++h) {
        float v = he[tid * LDST + h] + s1[tid * LDST + h] - mu;
        var += v * v;
      }
      var *= (1.f / HH);
      float rs = rsqrtf(var + EPSL);
      for (int h = 0; h < HH; ++h) {
        float v = he[tid * LDST + h] + s1[tid * LDST + h] - mu;
        he[tid * LDST + h] = v * rs * L.lneg[h] + L.lneb[h];
      }
    }
    __syncthreads();
  }

  // ---- Stage 4: score_x = silu(hn@Wsx1 + b) @ Wsx2 + b ----
  for (int nt = 0; nt < 8; ++nt) {
    int col = nt * 16 + l16;
    v8f acc; float bb = p.b_sx1[col];
#pragma unroll
    for (int r = 0; r < 8; ++r) acc[r] = bb;
    acc = wmma_span(hn, LDST, arow, p.w_sx1, HH, col, HH, half2, acc);
#pragma unroll
    for (int r = 0; r < 8; ++r) acc[r] = silu_f(acc[r]);
    store_tile(s1, LDST, mt, col, half, acc);
  }
  __syncthreads();
  for (int nt = 0; nt < 4; ++nt) {     // N = D = 64
    int col = nt * 16 + l16;
    v8f acc; float bb = p.b_sx2[col];
#pragma unroll
    for (int r = 0; r < 8; ++r) acc[r] = bb;
    acc = wmma_span(s1, LDST, arow, p.w_sx2, DD, col, HH, half2, acc);
    store_tile(p.out_x + b * NN * DD, DD, mt, col, half, acc);
  }

  // ---- Stage 5: u = hn @ W_sa1 (bias folded into score_a kernel) -> workspace ----
  for (int nt = 0; nt < 8; ++nt) {
    int col = nt * 16 + l16;
    v8f acc;
#pragma unroll
    for (int r = 0; r < 8; ++r) acc[r] = 0.f;
    acc = wmma_span(hn, LDST, arow, p.w_sa1, HH, col, HH, half2, acc);
    store_tile(p.u + b * NN * HH, HH, mt, col, half, acc);
  }
}

// score_a[b,i,j] = sum_h silu(0.5*(u_i[h]+u_j[h]) + b_sa1[h]) * w_sa2[h] + b_sa2
// (pre-activation is symmetric in (i,j), so (s + s^T)/2 == s; zero the diagonal)
__global__ __launch_bounds__(128) void gnn_score_a(const float* __restrict__ u,
                                                   const float* __restrict__ bsa1,
                                                   const float* __restrict__ wsa2,
                                                   const float* __restrict__ bsa2,
                                                   float* __restrict__ out_a) {
  __shared__ float ui[HH], bb[HH], w2[HH];
  int bi = blockIdx.x;
  int b = bi >> 7, i = bi & 127;
  int tid = threadIdx.x;
  const float* ub = u + b * NN * HH;
  ui[tid] = ub[i * HH + tid];
  bb[tid] = bsa1[tid];
  w2[tid] = wsa2[tid];
  __syncthreads();
  int j = tid;
  const float* uj = ub + j * HH;
  float s = bsa2[0];
  for (int h = 0; h < HH; ++h) {
    float xv = 0.5f * (ui[h] + uj[h]) + bb[h];
    s += silu_f(xv) * w2[h];
  }
  if (j == i) s = 0.f;
  out_a[b * NN * NN + i * NN + j] = s;
}

extern "C" void kernel_launch(void* const* d_in, const int* in_sizes, int n_in,
                              void* d_out, int out_size, void* d_ws, size_t ws_size,
                              hipStream_t stream) {
  (void)in_sizes; (void)n_in; (void)out_size; (void)ws_size;
  NetP p;
  p.x    = (const float*)d_in[0];
  p.a    = (const float*)d_in[1];
  p.t    = (const float*)d_in[2];
  p.w_t1 = (const float*)d_in[3];  p.b_t1 = (const float*)d_in[4];
  p.w_t2 = (const float*)d_in[5];  p.b_t2 = (const float*)d_in[6];
  p.w_np = (const float*)d_in[7];  p.b_np = (const float*)d_in[8];
  p.w_ap = (const float*)d_in[9];  p.b_ap = (const float*)d_in[10];
  p.w_sx1 = (const float*)d_in[11]; p.b_sx1 = (const float*)d_in[12];
  p.w_sx2 = (const float*)d_in[13]; p.b_sx2 = (const float*)d_in[14];
  p.w_sa1 = (const float*)d_in[15]; p.b_sa1 = (const float*)d_in[16];
  p.w_sa2 = (const float*)d_in[17]; p.b_sa2 = (const float*)d_in[18];
  for (int l = 0; l < 4; ++l) {
    int base = 19 + l * 10;
    p.lyr[l].wmsg = (const float*)d_in[base + 0];
    p.lyr[l].bmsg = (const float*)d_in[base + 1];
    p.lyr[l].wupd = (const float*)d_in[base + 2];
    p.lyr[l].bupd = (const float*)d_in[base + 3];
    p.lyr[l].wedg = (const float*)d_in[base + 4];
    p.lyr[l].bedg = (const float*)d_in[base + 5];
    p.lyr[l].lnng = (const float*)d_in[base + 6];
    p.lyr[l].lnnb = (const float*)d_in[base + 7];
    p.lyr[l].lneg = (const float*)d_in[base + 8];
    p.lyr[l].lneb = (const float*)d_in[base + 9];
  }
  float* out = (float*)d_out;
  p.out_x = out;
  p.u     = (float*)d_ws;

  gnn_main<<<8, 256, 0, stream>>>(p);
  gnn_score_a<<<8 * 128, 128, 0, stream>>>(p.u, p.b_sa1, p.w_sa2, p.b_sa2,
                                           out + 8 * 128 * 64);
}